// GINEncoder_Global_75024488726862
// MI455X (gfx1250) — compile-verified
//
#include <hip/hip_runtime.h>

#define HDIM 128
#define BM   32
#define LDA  132   // padded row so strided A-fragment reads avoid bank conflicts

typedef float v2f __attribute__((ext_vector_type(2)));
typedef float v8f __attribute__((ext_vector_type(8)));

// ---------------------------------------------------------------------------
// Generic fused GEMM: out[M,128] = epi( (A [+A2]) @ Wt[128,128] + bias )
//   epi: optional ReLU, optional per-row scale (cutoff mask), optional
//        residual add (may alias out: same element read->write by same thread).
// Wave32 WMMA f32 16x16x4 per CDNA5 ISA 7.12.2 layouts.
// ---------------------------------------------------------------------------
__global__ __launch_bounds__(256)
void k_gemm128(const float* __restrict__ A, const float* __restrict__ A2,
               const float* __restrict__ Wt, const float* __restrict__ bias,
               const float* __restrict__ rowscale, const float* __restrict__ res,
               float* __restrict__ out, int M, int relu_out)
{
    __shared__ float sW[64][HDIM];   // one K-half of the 128x128 weight (32 KB)
    __shared__ float sA[BM][LDA];    // 32x128 A tile, padded (16.9 KB)

    const int tid = threadIdx.x;
    const int m0  = blockIdx.x * BM;

    // ---- stage A (+A2) tile into LDS (float4, coalesced) ----
    for (int i = tid * 4; i < BM * HDIM; i += 256 * 4) {
        const int r = i >> 7, c = i & 127;
        float4 a4 = make_float4(0.f, 0.f, 0.f, 0.f);
        if (m0 + r < M) {
            a4 = *(const float4*)(A + (size_t)(m0 + r) * HDIM + c);
            if (A2) {
                const float4 b4 = *(const float4*)(A2 + (size_t)(m0 + r) * HDIM + c);
                a4.x += b4.x; a4.y += b4.y; a4.z += b4.z; a4.w += b4.w;
            }
        }
        *(float4*)(&sA[r][c]) = a4;
    }

    const int wave  = tid >> 5;        // 0..7 -> 16-wide N tile
    const int lane  = tid & 31;
    const int lhalf = lane >> 4;       // K-pair selector per ISA A/B layout
    const int lrow  = lane & 15;
    const int n0    = wave * 16;

    const float bval = bias[n0 + lrow];
    v8f acc0, acc1;
    #pragma unroll
    for (int v = 0; v < 8; ++v) { acc0[v] = bval; acc1[v] = bval; }

    for (int kb = 0; kb < HDIM; kb += 64) {
        __syncthreads();   // A ready (iter 0) / previous sW consumed (iter 1)
        for (int i = tid * 4; i < 64 * HDIM; i += 256 * 4) {
            const float4 w4 = *(const float4*)(Wt + (size_t)kb * HDIM + i);
            *(float4*)(&sW[i >> 7][i & 127]) = w4;
        }
        __syncthreads();

        #pragma unroll
        for (int k = 0; k < 64; k += 4) {
            const int kk = k + 2 * lhalf;
            v2f b, a0, a1;
            b[0]  = sW[kk][n0 + lrow];
            b[1]  = sW[kk + 1][n0 + lrow];
            a0[0] = sA[lrow][kb + kk];
            a0[1] = sA[lrow][kb + kk + 1];
            a1[0] = sA[16 + lrow][kb + kk];
            a1[1] = sA[16 + lrow][kb + kk + 1];
            acc0 = __builtin_amdgcn_wmma_f32_16x16x4_f32(
                       false, a0, false, b, (short)0, acc0, false, false);
            acc1 = __builtin_amdgcn_wmma_f32_16x16x4_f32(
                       false, a1, false, b, (short)0, acc1, false, false);
        }
    }

    // ---- epilogue: C/D layout -> VGPR v = row v (lanes 0-15) / v+8 (16-31)
    const int n = n0 + lrow;
    #pragma unroll
    for (int sub = 0; sub < 2; ++sub) {
        v8f acc = sub ? acc1 : acc0;
        #pragma unroll
        for (int v = 0; v < 8; ++v) {
            const int m = m0 + sub * 16 + lhalf * 8 + v;
            if (m < M) {
                float val = acc[v];
                if (relu_out) val = fmaxf(val, 0.f);
                if (rowscale) val *= rowscale[m];
                if (res)      val += res[(size_t)m * HDIM + n];
                out[(size_t)m * HDIM + n] = val;
            }
        }
    }
}

// ---------------------------------------------------------------------------
// hidden = relu(z @ w0a + b0a): K=5, too small for WMMA -> scalar, coalesced
// ---------------------------------------------------------------------------
__global__ void k_node_hidden(const float* __restrict__ z, const float* __restrict__ w0a,
                              const float* __restrict__ b0a, float* __restrict__ outp, int N_)
{
    const int idx = blockIdx.x * blockDim.x + threadIdx.x;
    if (idx >= N_ * HDIM) return;
    const int n = idx >> 7, j = idx & 127;
    float s = b0a[j];
    #pragma unroll
    for (int k = 0; k < 5; ++k) s += z[n * 5 + k] * w0a[k * HDIM + j];
    outp[idx] = fmaxf(s, 0.f);
}

__global__ void k_edge_mask(const float* __restrict__ elen, float* __restrict__ mask, int E_)
{
    const int e = blockIdx.x * blockDim.x + threadIdx.x;
    if (e < E_) mask[e] = (elen[e] <= 10.0f) ? 1.0f : 0.0f;
}

__global__ void k_zero(float* __restrict__ p, int n)
{
    const int i = blockIdx.x * blockDim.x + threadIdx.x;
    if (i < n) p[i] = 0.f;
}

// ---------------------------------------------------------------------------
// m = relu(h[src] + Wedge); agg[dst] += m   (float4 loads, f32 atomics)
// ---------------------------------------------------------------------------
__global__ void k_scatter(const float* __restrict__ h, const float* __restrict__ Wedge,
                          const int* __restrict__ src, const int* __restrict__ dst,
                          float* __restrict__ agg, int E_)
{
    const int idx = blockIdx.x * blockDim.x + threadIdx.x;  // E*32 threads
    if (idx >= E_ * 32) return;
    const int e = idx >> 5;
    const int j = (idx & 31) * 4;
    const int s = src[e], d = dst[e];
    const float4 hv = *(const float4*)(h + (size_t)s * HDIM + j);
    const float4 wv = *(const float4*)(Wedge + (size_t)e * HDIM + j);
    float* p = agg + (size_t)d * HDIM + j;
    atomicAdd(p + 0, fmaxf(hv.x + wv.x, 0.f));
    atomicAdd(p + 1, fmaxf(hv.y + wv.y, 0.f));
    atomicAdd(p + 2, fmaxf(hv.z + wv.z, 0.f));
    atomicAdd(p + 3, fmaxf(hv.w + wv.w, 0.f));
}

extern "C" void kernel_launch(void* const* d_in, const int* in_sizes, int n_in,
                              void* d_out, int out_size, void* d_ws, size_t ws_size,
                              hipStream_t stream)
{
    (void)n_in; (void)out_size; (void)ws_size;
    const float* z     = (const float*)d_in[0];
    const int*   eidx  = (const int*)  d_in[1];
    const float* eattr = (const float*)d_in[2];
    const float* elen  = (const float*)d_in[3];
    const float* w0a = (const float*)d_in[4];  const float* b0a = (const float*)d_in[5];
    const float* w0b = (const float*)d_in[6];  const float* b0b = (const float*)d_in[7];
    const float* w1a = (const float*)d_in[8];  const float* b1a = (const float*)d_in[9];
    const float* w1b = (const float*)d_in[10]; const float* b1b = (const float*)d_in[11];
    const float* w2a = (const float*)d_in[12]; const float* b2a = (const float*)d_in[13];
    const float* w2b = (const float*)d_in[14]; const float* b2b = (const float*)d_in[15];

    const int N_ = in_sizes[0] / 5;   // z is [N,5]
    const int E_ = in_sizes[3];       // edge_length is [E]

    float* h       = (float*)d_out;                       // [N,128] running node state
    float* Wedge   = (float*)d_ws;                        // [E,128] edge gate
    float* scratch = Wedge + (size_t)E_ * HDIM;           // [E,128] reusable
    float* mask    = scratch + (size_t)E_ * HDIM;         // [E]
    float* agg     = scratch;                             // [N,128] (inside scratch)
    float* hid     = scratch + (size_t)N_ * HDIM;         // [N,128] (inside scratch)

    const int* src = eidx;
    const int* dst = eidx + E_;

    const int gN32 = (N_ + BM - 1) / BM;
    const int gE32 = (E_ + BM - 1) / BM;

    // cutoff mask
    k_edge_mask<<<(E_ + 255) / 256, 256, 0, stream>>>(elen, mask, E_);
    // x = mlp(z):  hidden (scalar, K=5) then WMMA GEMM -> h
    k_node_hidden<<<(N_ * HDIM + 255) / 256, 256, 0, stream>>>(z, w0a, b0a, scratch, N_);
    k_gemm128<<<gN32, 256, 0, stream>>>(scratch, nullptr, w0b, b0b, nullptr, nullptr, h, N_, 0);
    // W = mlp(edge_attr) * cutoff_mask  (two edge-sized WMMA GEMMs)
    k_gemm128<<<gE32, 256, 0, stream>>>(eattr,   nullptr, w2a, b2a, nullptr, nullptr, scratch, E_, 1);
    k_gemm128<<<gE32, 256, 0, stream>>>(scratch, nullptr, w2b, b2b, mask,    nullptr, Wedge,   E_, 0);

    for (int conv = 0; conv < 3; ++conv) {
        k_zero<<<(N_ * HDIM + 255) / 256, 256, 0, stream>>>(agg, N_ * HDIM);
        k_scatter<<<(E_ * 32 + 255) / 256, 256, 0, stream>>>(h, Wedge, src, dst, agg, E_);
        // mlp(agg + h): layer 1 (fused add via A2, ReLU out)
        k_gemm128<<<gN32, 256, 0, stream>>>(agg, h, w1a, b1a, nullptr, nullptr, hid, N_, 1);
        // layer 2: optional inter-conv ReLU, fused residual h = out + h (in-place)
        k_gemm128<<<gN32, 256, 0, stream>>>(hid, nullptr, w1b, b1b, nullptr, h, h, N_, (conv < 2) ? 1 : 0);
    }
}